// AgentCollisionLoss_34522947125766
// MI455X (gfx1250) — compile-verified
//
#include <hip/hip_runtime.h>
#include <math.h>

// Problem constants (fixed by setup_inputs)
#define B_ 32
#define N_ 16
#define T_ 52
#define D_ 5
#define BUFFER_DIST 0.2f
#define SPEED_TH 0.5f
// log2(0.9)
#define LOG2_DECAY (-0.15200309344504997f)

typedef __attribute__((ext_vector_type(2))) float v2f;
typedef __attribute__((ext_vector_type(8))) float v8f;

// world_cent layout: t fastest -> coalesced reads when threads map to t
__device__ __forceinline__ int wc_idx(int b, int n, int d, int xy, int t) {
    return ((((b * N_ + n) * D_ + d) * 2 + xy) * T_ + t);
}

// ---------------------------------------------------------------------------
// Kernel 1: agent-frame -> world-frame disk centroids.
// One thread per (b,n,t); writes 5 disks * (x,y).
// ---------------------------------------------------------------------------
__global__ void k1_world_cent(const float* __restrict__ x,
                              const float* __restrict__ extent,
                              const float* __restrict__ wfa,
                              float* __restrict__ wc) {
    int g = blockIdx.x * blockDim.x + threadIdx.x;
    if (g >= B_ * N_ * T_) return;
    int t = g % T_;
    int n = (g / T_) % N_;
    int b = g / (T_ * N_);

    const float* xp = x + ((size_t)((b * N_ + n) * T_ + t)) * 6;
    float px = xp[0], py = xp[1], yaw = xp[3];

    const float* w = wfa + b * 9;
    float R00 = w[0], R01 = w[1], tx = w[2];
    float R10 = w[3], R11 = w[4], ty = w[5];

    // pos_g[e] = R[e,0]*px + R[e,1]*py + t[e]
    float gx = R00 * px + R01 * py + tx;
    float gy = R10 * px + R11 * py + ty;

    float yaw_g = yaw + atan2f(R10, R00);
    float s, c;
    sincosf(yaw_g, &s, &c);

    float L   = extent[b * 3 + 0];
    float rad = extent[b * 3 + 1] * 0.5f;
    float cmin = rad - 0.5f * L;
    float cmax = 0.5f * L - rad;
    float step = (cmax - cmin) * 0.25f;   // linspace(0,1,5) spacing

    // world_cent = cent_x * [c, s] + pos_g  (centroids have y==0)
#pragma unroll
    for (int d = 0; d < D_; ++d) {
        float cx = cmin + step * (float)d;
        wc[wc_idx(b, n, d, 0, t)] = cx * c + gx;
        wc[wc_idx(b, n, d, 1, t)] = cx * s + gy;
    }
}

// ---------------------------------------------------------------------------
// Kernel 2: per (i,n,t): sum over agents j (same scene, j!=i) of hinge
// penalty on min disk-pair distance. One block per (i,n), threads = t.
// Writes S[(i*16+n)*52 + t].
// ---------------------------------------------------------------------------
__global__ void k2_pair_penalty(const float* __restrict__ wc,
                                const float* __restrict__ extent,
                                const int* __restrict__ scene,
                                float* __restrict__ S) {
    int m = blockIdx.x;        // 0..511 = i*16 + n
    int i = m >> 4;
    int n = m & 15;
    int t = threadIdx.x;
    if (t >= T_) return;

    float xi[D_], yi[D_];
#pragma unroll
    for (int d = 0; d < D_; ++d) {
        xi[d] = wc[wc_idx(i, n, d, 0, t)];
        yi[d] = wc[wc_idx(i, n, d, 1, t)];
    }
    float radi = extent[i * 3 + 1] * 0.5f;
    int   si   = scene[i];

    float acc = 0.0f;
    for (int j = 0; j < B_; ++j) {
        if (j == i || scene[j] != si) continue;
        float minq = 3.4e38f;
#pragma unroll
        for (int d2 = 0; d2 < D_; ++d2) {
            float xj = wc[wc_idx(j, n, d2, 0, t)];
            float yj = wc[wc_idx(j, n, d2, 1, t)];
#pragma unroll
            for (int d1 = 0; d1 < D_; ++d1) {
                float dx = xi[d1] - xj;
                float dy = yi[d1] - yj;
                minq = fminf(minq, dx * dx + dy * dy);
            }
        }
        float dist = sqrtf(minq);                       // min(sqrt) == sqrt(min)
        float pd   = radi + extent[j * 3 + 1] * 0.5f + BUFFER_DIST;
        if (dist <= pd) acc += 1.0f - dist / pd;
    }
    S[m * T_ + t] = acc;
}

// ---------------------------------------------------------------------------
// Kernel 3: weighted time reduction via V_WMMA_F32_16X16X4_F32.
// loss[m] = (1/B) * sum_t S[m,t] * w[t],  w[t] = 0.9^t / sum(0.9^t)
// One wave (32 threads) per 16 rows: A = 16x4 tile of S (f32, 2 VGPRs:
// VGPR0 = K0|K2, VGPR1 = K1|K3 across lane halves), B = 4x16 with w
// replicated across all 16 columns, f32 C accumulated over 13 K-steps.
// Column 0 of C holds the answer (all columns identical).
// ---------------------------------------------------------------------------
__global__ void k3_wmma_reduce(const float* __restrict__ S,
                               const float* __restrict__ curr_speed,
                               float* __restrict__ out) {
    int mt   = blockIdx.x;          // 0..31, rows [mt*16, mt*16+16)
    int lane = threadIdx.x;         // 0..31 (wave32)
    int row  = mt * 16 + (lane & 15);
    int koff = (lane < 16) ? 0 : 2; // A/B layout: lane halves hold K, K+2

    float norm = 0.1f / (1.0f - exp2f(52.0f * LOG2_DECAY)); // closed-form sum

    v8f c = {0.f, 0.f, 0.f, 0.f, 0.f, 0.f, 0.f, 0.f};
    for (int k = 0; k < 13; ++k) {
        int tb = 4 * k + koff;
        v2f a, b;
        a.x = S[row * T_ + tb];
        a.y = S[row * T_ + tb + 1];
        b.x = exp2f((float)tb * LOG2_DECAY) * norm;
        b.y = exp2f((float)(tb + 1) * LOG2_DECAY) * norm;
        // D = A(16x4) * B(4x16) + C
        c = __builtin_amdgcn_wmma_f32_16x16x4_f32(
                /*neg_a=*/false, a, /*neg_b=*/false, b,
                /*c_mod=*/(short)0, c, /*reuse_a=*/false, /*reuse_b=*/false);
    }

    // Extract column N==0: lane 0 holds M=0..7 (VGPR 0..7), lane 16 holds M=8..15
    if ((lane & 15) == 0) {
        int baseM = mt * 16 + ((lane < 16) ? 0 : 8);
#pragma unroll
        for (int r = 0; r < 8; ++r) {
            int mrow = baseM + r;
            int ii   = mrow >> 4;
            bool moving = fabsf(curr_speed[ii]) > SPEED_TH;
            float v = c[r] * (1.0f / (float)B_);
            out[mrow] = moving ? v : 0.0f;
        }
    }
}

// ---------------------------------------------------------------------------
extern "C" void kernel_launch(void* const* d_in, const int* in_sizes, int n_in,
                              void* d_out, int out_size, void* d_ws, size_t ws_size,
                              hipStream_t stream) {
    const float* x      = (const float*)d_in[0];  // (32,16,52,6)
    const float* extent = (const float*)d_in[1];  // (32,3)
    const float* wfa    = (const float*)d_in[2];  // (32,3,3)
    const float* speed  = (const float*)d_in[3];  // (32,)
    const int*   scene  = (const int*)d_in[4];    // (32,)
    float* out = (float*)d_out;                   // (32,16)

    float* wc = (float*)d_ws;                                   // 266,240 f32
    float* S  = wc + (size_t)B_ * N_ * D_ * 2 * T_;             // 26,624 f32

    // Phase 1: 26,624 threads
    k1_world_cent<<<(B_ * N_ * T_ + 255) / 256, 256, 0, stream>>>(x, extent, wfa, wc);
    // Phase 2: one block per (i,n), 64 threads (52 active over t)
    k2_pair_penalty<<<B_ * N_, 64, 0, stream>>>(wc, extent, scene, S);
    // Phase 3: 32 waves, one per 16 output rows, WMMA f32 16x16x4
    k3_wmma_reduce<<<32, 32, 0, stream>>>(S, speed, out);
}